// EdgeGATLayer_53936199303551
// MI455X (gfx1250) — compile-verified
//
#include <hip/hip_runtime.h>

#define NN 50000
#define EE 800000
#define ND 256      // NODE_DIM
#define ED 64       // EDGE_DIM
#define HID 128
#define HEADS 8
#define HD 16       // HEAD_DIM

typedef __attribute__((ext_vector_type(2))) float v2f;
typedef __attribute__((ext_vector_type(8))) float v8f;

#if defined(__HIP_DEVICE_COMPILE__) && __has_builtin(__builtin_amdgcn_global_load_async_to_lds_b128)
#define HAS_ASYNC 1
#else
#define HAS_ASYNC 0
#endif

#if HAS_ASYNC
typedef int vsi4 __attribute__((__vector_size__(4 * sizeof(int))));
typedef __attribute__((address_space(1))) vsi4 g_vsi4;   // global int4
typedef __attribute__((address_space(3))) vsi4 l_vsi4;   // LDS int4
#endif

// ---------------------------------------------------------------- helpers
// stage one 16-byte chunk global -> LDS (async-to-LDS on gfx1250, else via VGPRs)
__device__ __forceinline__ void stage16(const float* g, float* l) {
#if HAS_ASYNC
    __builtin_amdgcn_global_load_async_to_lds_b128(
        (g_vsi4*)(float*)g, (l_vsi4*)l, 0, 0);
#else
    *reinterpret_cast<float4*>(l) = *reinterpret_cast<const float4*>(g);
#endif
}

__device__ __forceinline__ void stage_wait() {
#if HAS_ASYNC
#if __has_builtin(__builtin_amdgcn_s_wait_asynccnt)
    __builtin_amdgcn_s_wait_asynccnt(0);
#else
    asm volatile("s_wait_asynccnt 0x0" ::: "memory");
#endif
#endif
}

__device__ __forceinline__ void atomicMaxF(float* addr, float val) {
    unsigned* ua = (unsigned*)addr;
    unsigned old = *ua;
    while (true) {
        float f = __uint_as_float(old);
        if (f >= val) break;
        unsigned assumed = old;
        old = atomicCAS(ua, assumed, __float_as_uint(val));
        if (old == assumed) break;
    }
}

// ---------------------------------------------------------------- init
__global__ void fill_neg_kernel(float* __restrict__ p, int n) {
    int i = blockIdx.x * 256 + threadIdx.x;
    if (i < n) p[i] = -3.0e38f;
}

// w_ell[h][k] = sum_d att_edge[h][d] * W_edge[h*16+d][k]   (8 x 64)
__global__ void well_kernel(const float* __restrict__ W_edge,
                            const float* __restrict__ att_edge,
                            float* __restrict__ w_ell) {
    int t = blockIdx.x * 256 + threadIdx.x;
    if (t >= HEADS * ED) return;
    int h = t / ED, k = t % ED;
    float acc = 0.f;
    for (int d = 0; d < HD; ++d)
        acc += att_edge[h * HD + d] * W_edge[(h * HD + d) * ED + k];
    w_ell[t] = acc;
}

// ---------------------------------------------------------------- node GEMM (WMMA f32)
// h = X @ Wn^T,  res = X @ Wr^T ; 16-node strip per block, wave j = head j cols
__global__ __launch_bounds__(256) void node_gemm_kernel(
        const float* __restrict__ X, const float* __restrict__ Wn,
        const float* __restrict__ Wr, float* __restrict__ h,
        float* __restrict__ res) {
    __shared__ float xt[16][260];              // pad 4: conflict-free, rows 16B-aligned
    const int nbase = blockIdx.x * 16;
    const int t = threadIdx.x;
    for (int i = 0; i < 4; ++i) {              // stage 16x256 fp32 = 1024 x 16B
        int idx = t + i * 256;
        int row = idx >> 6, c4 = idx & 63;
        stage16(X + (size_t)(nbase + row) * ND + c4 * 4, &xt[row][c4 * 4]);
    }
    stage_wait();
    __syncthreads();
    const int wave = t >> 5, lane = t & 31;
    const int rowA = lane & 15;
    const int kk   = (lane >> 4) * 2;          // A layout: lanes>=16 hold K=2,3
    const int bcol = wave * 16 + (lane & 15);  // B layout: lane holds column N
    const float* wnrow = Wn + (size_t)bcol * ND;
    const float* wrrow = Wr + (size_t)bcol * ND;
    v8f acc_h = {0,0,0,0,0,0,0,0};
    v8f acc_r = {0,0,0,0,0,0,0,0};
    for (int k0 = 0; k0 < ND; k0 += 4) {
        v2f a  = *reinterpret_cast<const v2f*>(&xt[rowA][k0 + kk]);
        v2f bh = *reinterpret_cast<const v2f*>(wnrow + k0 + kk);
        v2f br = *reinterpret_cast<const v2f*>(wrrow + k0 + kk);
        acc_h = __builtin_amdgcn_wmma_f32_16x16x4_f32(false, a, false, bh,
                                                      (short)0, acc_h, false, false);
        acc_r = __builtin_amdgcn_wmma_f32_16x16x4_f32(false, a, false, br,
                                                      (short)0, acc_r, false, false);
    }
    const int colOut = wave * 16 + (lane & 15);
    const int mofs   = (lane >> 4) * 8;        // C layout: lanes>=16 hold rows 8..15
    for (int v = 0; v < 8; ++v) {
        int m = v + mofs;
        h  [(size_t)(nbase + m) * HID + colOut] = acc_h[v];
        res[(size_t)(nbase + m) * HID + colOut] = acc_r[v];
    }
}

// ---------------------------------------------------------------- per-node attn dots
__global__ void adot_kernel(const float* __restrict__ h,
                            const float* __restrict__ att_src,
                            const float* __restrict__ att_dst,
                            float* __restrict__ adot_s,
                            float* __restrict__ adot_d) {
    int t = blockIdx.x * 256 + threadIdx.x;
    if (t >= NN * HEADS) return;
    int n = t >> 3, hh = t & 7;
    const float* hp = h + (size_t)n * HID + hh * HD;
    float s = 0.f, d = 0.f;
    for (int i = 0; i < HD; ++i) {
        float v = hp[i];
        s += v * att_src[hh * HD + i];
        d += v * att_dst[hh * HD + i];
    }
    adot_s[t] = s;
    adot_d[t] = d;
}

// ---------------------------------------------------------------- edge logits + seg-max
__global__ __launch_bounds__(256) void edge_logits_kernel(
        const float* __restrict__ EA, const int* __restrict__ ei,
        const float* __restrict__ w_ell, const float* __restrict__ adot_s,
        const float* __restrict__ adot_d, float* __restrict__ logits,
        float* __restrict__ mbuf) {
    __shared__ float ea[32][68];
    __shared__ float wl[8][68];
    const int ebase = blockIdx.x * 32;
    const int t = threadIdx.x;
    for (int i = 0; i < 2; ++i) {              // 32x64 fp32 = 512 x 16B
        int idx = t + i * 256;
        int row = idx >> 4, c4 = idx & 15;
        stage16(EA + (size_t)(ebase + row) * ED + c4 * 4, &ea[row][c4 * 4]);
    }
    for (int i = t; i < HEADS * ED; i += 256)
        wl[i >> 6][i & 63] = w_ell[i];
    stage_wait();
    __syncthreads();
    const int el = t >> 3, hh = t & 7;
    const int eidx = ebase + el;
    float acc = 0.f;
    for (int k = 0; k < ED; ++k) acc += ea[el][k] * wl[hh][k];
    int src = ei[eidx], dst = ei[EE + eidx];
    float lg = acc + adot_s[src * HEADS + hh] + adot_d[dst * HEADS + hh];
    lg = lg > 0.f ? lg : 0.2f * lg;            // leaky_relu 0.2
    logits[(size_t)eidx * HEADS + hh] = lg;
    atomicMaxF(&mbuf[dst * HEADS + hh], lg);
}

// ---------------------------------------------------------------- exp + seg-sum
__global__ void edge_exp_kernel(const int* __restrict__ ei,
                                float* __restrict__ logits,
                                const float* __restrict__ mbuf,
                                float* __restrict__ sbuf) {
    int t = blockIdx.x * 256 + threadIdx.x;    // E*HEADS = 6.4M
    int eidx = t >> 3, hh = t & 7;
    int dst = ei[EE + eidx];
    float ex = expf(logits[t] - mbuf[dst * HEADS + hh]);
    logits[t] = ex;
    atomicAdd(&sbuf[dst * HEADS + hh], ex);
}

// ---------------------------------------------------------------- edge GEMM + scatter (WMMA f32)
__global__ __launch_bounds__(256) void edge_scatter_kernel(
        const float* __restrict__ EA, const float* __restrict__ We,
        const int* __restrict__ ei, const float* __restrict__ h,
        const float* __restrict__ exv, const float* __restrict__ sbuf,
        float* __restrict__ outb) {
    __shared__ float ea[16][68];
    const int ebase = blockIdx.x * 16;
    const int t = threadIdx.x;
    {                                           // 16x64 fp32 = 256 x 16B
        int row = t >> 4, c4 = t & 15;
        stage16(EA + (size_t)(ebase + row) * ED + c4 * 4, &ea[row][c4 * 4]);
    }
    stage_wait();
    __syncthreads();
    const int wave = t >> 5, lane = t & 31;
    const int rowA = lane & 15;
    const int kk   = (lane >> 4) * 2;
    const int bcol = wave * 16 + (lane & 15);
    const float* werow = We + (size_t)bcol * ED;
    v8f acc = {0,0,0,0,0,0,0,0};
    for (int k0 = 0; k0 < ED; k0 += 4) {
        v2f a = *reinterpret_cast<const v2f*>(&ea[rowA][k0 + kk]);
        v2f b = *reinterpret_cast<const v2f*>(werow + k0 + kk);
        acc = __builtin_amdgcn_wmma_f32_16x16x4_f32(false, a, false, b,
                                                    (short)0, acc, false, false);
    }
    const int d    = lane & 15;                 // column (head-dim index) from C layout
    const int mofs = (lane >> 4) * 8;
    for (int v = 0; v < 8; ++v) {
        int eidx = ebase + v + mofs;            // uniform per 16-lane half
        int src = ei[eidx], dst = ei[EE + eidx];
        float alpha = exv[(size_t)eidx * HEADS + wave] / sbuf[dst * HEADS + wave];
        float val = alpha * (h[(size_t)src * HID + wave * HD + d] + acc[v]);
        atomicAdd(&outb[(size_t)dst * HID + wave * HD + d], val);
    }
}

// ---------------------------------------------------------------- residual + LN + ELU
__global__ __launch_bounds__(256) void finalize_kernel(
        const float* __restrict__ outb, const float* __restrict__ resb,
        const float* __restrict__ gamma, const float* __restrict__ beta,
        float* __restrict__ y) {
    const int t = threadIdx.x;
    const int wave = t >> 5, lane = t & 31;
    const int n = blockIdx.x * 8 + wave;        // one wave per node, 4 cols/lane
    const size_t base = (size_t)n * HID + lane * 4;
    float4 a = *reinterpret_cast<const float4*>(outb + base);
    float4 b = *reinterpret_cast<const float4*>(resb + base);
    float x0 = a.x + b.x, x1 = a.y + b.y, x2 = a.z + b.z, x3 = a.w + b.w;
    float s1 = x0 + x1 + x2 + x3;
    float s2 = x0*x0 + x1*x1 + x2*x2 + x3*x3;
    for (int m = 1; m < 32; m <<= 1) {
        s1 += __shfl_xor(s1, m, 32);
        s2 += __shfl_xor(s2, m, 32);
    }
    float mu  = s1 * (1.0f / HID);
    float var = s2 * (1.0f / HID) - mu * mu;
    float inv = rsqrtf(var + 1e-5f);
    float4 g  = *reinterpret_cast<const float4*>(gamma + lane * 4);
    float4 be = *reinterpret_cast<const float4*>(beta  + lane * 4);
    float o0 = (x0 - mu) * inv * g.x + be.x;
    float o1 = (x1 - mu) * inv * g.y + be.y;
    float o2 = (x2 - mu) * inv * g.z + be.z;
    float o3 = (x3 - mu) * inv * g.w + be.w;
    o0 = o0 > 0.f ? o0 : expf(o0) - 1.f;
    o1 = o1 > 0.f ? o1 : expf(o1) - 1.f;
    o2 = o2 > 0.f ? o2 : expf(o2) - 1.f;
    o3 = o3 > 0.f ? o3 : expf(o3) - 1.f;
    float4 o = {o0, o1, o2, o3};
    *reinterpret_cast<float4*>(y + base) = o;
}

// ---------------------------------------------------------------- launch
extern "C" void kernel_launch(void* const* d_in, const int* in_sizes, int n_in,
                              void* d_out, int out_size, void* d_ws, size_t ws_size,
                              hipStream_t stream) {
    const float* node_feat = (const float*)d_in[0];
    const int*   ei        = (const int*)  d_in[1];   // [2,E] int32
    const float* edge_attr = (const float*)d_in[2];
    const float* W_node    = (const float*)d_in[3];
    const float* W_edge    = (const float*)d_in[4];
    const float* att_src   = (const float*)d_in[5];
    const float* att_dst   = (const float*)d_in[6];
    const float* att_edge  = (const float*)d_in[7];
    const float* W_res     = (const float*)d_in[8];
    const float* ln_scale  = (const float*)d_in[9];
    const float* ln_bias   = (const float*)d_in[10];

    // workspace layout (~109 MB of fp32)
    float* ws     = (float*)d_ws;
    float* h      = ws;                                // N*HID
    float* res    = h      + (size_t)NN * HID;         // N*HID
    float* adot_s = res    + (size_t)NN * HID;         // N*H
    float* adot_d = adot_s + (size_t)NN * HEADS;       // N*H
    float* w_ell  = adot_d + (size_t)NN * HEADS;       // 8*64
    float* logits = w_ell  + (size_t)HEADS * ED;       // E*H (reused for exp)
    float* mbuf   = logits + (size_t)EE * HEADS;       // N*H
    float* sbuf   = mbuf   + (size_t)NN * HEADS;       // N*H
    float* outb   = sbuf   + (size_t)NN * HEADS;       // N*HID

    (void)hipMemsetAsync(sbuf, 0, (size_t)NN * HEADS * sizeof(float), stream);
    (void)hipMemsetAsync(outb, 0, (size_t)NN * HID   * sizeof(float), stream);
    fill_neg_kernel<<<(NN * HEADS + 255) / 256, 256, 0, stream>>>(mbuf, NN * HEADS);
    well_kernel<<<2, 256, 0, stream>>>(W_edge, att_edge, w_ell);

    node_gemm_kernel<<<NN / 16, 256, 0, stream>>>(node_feat, W_node, W_res, h, res);
    adot_kernel<<<(NN * HEADS + 255) / 256, 256, 0, stream>>>(h, att_src, att_dst,
                                                              adot_s, adot_d);
    edge_logits_kernel<<<EE / 32, 256, 0, stream>>>(edge_attr, ei, w_ell,
                                                    adot_s, adot_d, logits, mbuf);
    edge_exp_kernel<<<(EE * HEADS) / 256, 256, 0, stream>>>(ei, logits, mbuf, sbuf);
    edge_scatter_kernel<<<EE / 16, 256, 0, stream>>>(edge_attr, W_edge, ei, h,
                                                     logits, sbuf, outb);
    finalize_kernel<<<NN / 8, 256, 0, stream>>>(outb, res, ln_scale, ln_bias,
                                                (float*)d_out);
}